// NodeUpdate_88502096101546
// MI455X (gfx1250) — compile-verified
//
#include <hip/hip_runtime.h>
#include <math.h>

typedef __attribute__((ext_vector_type(2))) float v2f;
typedef __attribute__((ext_vector_type(8))) float v8f;

#define N_NODES 50000
#define N_EDGES 800000
#define EP_EDGES (N_EDGES + N_NODES)   // with self loops
#define C_IN 64
#define C_OUT 64
#define HEADS 4
#define HC (HEADS * C_OUT)             // 256
#define NEG_SLOPE 0.2f
#define BN_EPS 1e-5f

// ---------------------------------------------------------------- utilities
__global__ void fill_kernel(float* __restrict__ p, float v, int n) {
    int i = blockIdx.x * blockDim.x + threadIdx.x;
    if (i < n) p[i] = v;
}

__device__ __forceinline__ void atomicMaxF(float* addr, float v) {
    // float max via signed/unsigned int monotonic mapping
    if (v >= 0.0f) atomicMax((int*)addr, __float_as_int(v));
    else           atomicMin((unsigned int*)addr, __float_as_uint(v));
}

__device__ __forceinline__ float lrelu(float v) {
    return v > 0.0f ? v : NEG_SLOPE * v;
}

// ---------------------------------------------------------------- WMMA GEMM
// xh[N, 256] = xin[N, 64] @ W[64, 256], f32 WMMA 16x16x4, one tile per wave.
__global__ void gemm_wmma_kernel(const float* __restrict__ X,
                                 const float* __restrict__ W,
                                 float* __restrict__ XH) {
    const int lane    = threadIdx.x & 31;
    const int wave    = threadIdx.x >> 5;
    const int rowTile = blockIdx.x;               // 0..3124 (50000/16)
    const int colTile = blockIdx.y * 8 + wave;    // 0..15   (256/16)
    const int row0 = rowTile * 16;
    const int col0 = colTile * 16;
    const int mn    = lane & 15;                  // M for A, N for B/D
    const int khalf = (lane >> 4) << 1;           // lanes 0-15 -> K{0,1}, 16-31 -> K{2,3}

    v8f acc = {};
    const float* Arow = X + (size_t)(row0 + mn) * C_IN;
    #pragma unroll
    for (int k0 = 0; k0 < C_IN; k0 += 4) {
        v2f a, b;
        a.x = Arow[k0 + khalf];
        a.y = Arow[k0 + khalf + 1];
        b.x = W[(size_t)(k0 + khalf)     * HC + col0 + mn];
        b.y = W[(size_t)(k0 + khalf + 1) * HC + col0 + mn];
        acc = __builtin_amdgcn_wmma_f32_16x16x4_f32(
            /*neg_a=*/false, a, /*neg_b=*/false, b,
            /*c_mod=*/(short)0, acc, /*reuse_a=*/false, /*reuse_b=*/false);
    }
    // D layout: VGPR i -> M = i (lanes 0-15) / i+8 (lanes 16-31), N = lane&15
    const int rbase = row0 + ((lane >> 4) << 3);
    #pragma unroll
    for (int i = 0; i < 8; ++i)
        XH[(size_t)(rbase + i) * HC + col0 + mn] = acc[i];
}

// ---------------------------------------------------------------- attention logits
// alsrc[n,h] = sum_c xh[n,h,c]*asrc[h,c]  (block of 256 = one node)
__global__ void al_kernel(const float* __restrict__ xh,
                          const float* __restrict__ asrc,
                          const float* __restrict__ adst,
                          float* __restrict__ alsrc,
                          float* __restrict__ aldst) {
    __shared__ float ss[256];
    __shared__ float sd[256];
    const int n = blockIdx.x;
    const int t = threadIdx.x;
    const float xv = xh[(size_t)n * HC + t];
    ss[t] = xv * asrc[t];
    sd[t] = xv * adst[t];
    __syncthreads();
    #pragma unroll
    for (int s = 32; s > 0; s >>= 1) {
        if ((t & 63) < s) { ss[t] += ss[t + s]; sd[t] += sd[t + s]; }
        __syncthreads();
    }
    if ((t & 63) == 0) {
        const int h = t >> 6;
        alsrc[n * HEADS + h] = ss[t];
        aldst[n * HEADS + h] = sd[t];
    }
}

// ---------------------------------------------------------------- edge passes
__device__ __forceinline__ void edge_sd(const int* __restrict__ ei, int e, int& s, int& d) {
    if (e < N_EDGES) { s = ei[e]; d = ei[N_EDGES + e]; }
    else             { s = d = e - N_EDGES; }          // self loop
}

__global__ void edge_max_kernel(const int* __restrict__ ei,
                                const float* __restrict__ alsrc,
                                const float* __restrict__ aldst,
                                float* __restrict__ segmax) {
    const int e = blockIdx.x * blockDim.x + threadIdx.x;
    if (e >= EP_EDGES) return;
    int s, d; edge_sd(ei, e, s, d);
    const float4 as = ((const float4*)alsrc)[s];
    const float4 ad = ((const float4*)aldst)[d];
    atomicMaxF(&segmax[d * HEADS + 0], lrelu(as.x + ad.x));
    atomicMaxF(&segmax[d * HEADS + 1], lrelu(as.y + ad.y));
    atomicMaxF(&segmax[d * HEADS + 2], lrelu(as.z + ad.z));
    atomicMaxF(&segmax[d * HEADS + 3], lrelu(as.w + ad.w));
}

__global__ void edge_exp_kernel(const int* __restrict__ ei,
                                const float* __restrict__ alsrc,
                                const float* __restrict__ aldst,
                                const float* __restrict__ segmax,
                                float* __restrict__ pbuf,
                                float* __restrict__ segsum) {
    const int e = blockIdx.x * blockDim.x + threadIdx.x;
    if (e >= EP_EDGES) return;
    int s, d; edge_sd(ei, e, s, d);
    const float4 as = ((const float4*)alsrc)[s];
    const float4 ad = ((const float4*)aldst)[d];
    const float4 m  = ((const float4*)segmax)[d];
    float4 p;
    p.x = __expf(lrelu(as.x + ad.x) - m.x);
    p.y = __expf(lrelu(as.y + ad.y) - m.y);
    p.z = __expf(lrelu(as.z + ad.z) - m.z);
    p.w = __expf(lrelu(as.w + ad.w) - m.w);
    ((float4*)pbuf)[e] = p;
    atomicAdd(&segsum[d * HEADS + 0], p.x);
    atomicAdd(&segsum[d * HEADS + 1], p.y);
    atomicAdd(&segsum[d * HEADS + 2], p.z);
    atomicAdd(&segsum[d * HEADS + 3], p.w);
}

// aggregation with fused head mean: y[d,c] += 0.25 * sum_h alpha[h]*xh[s,h,c]
// 64 threads per edge (4 edges per 256-thread block)
__global__ void edge_agg_kernel(const int* __restrict__ ei,
                                const float* __restrict__ pbuf,
                                const float* __restrict__ segsum,
                                const float* __restrict__ xh,
                                float* __restrict__ y) {
    const int t = threadIdx.x;
    const int c = t & 63;
    const int e = blockIdx.x * 4 + (t >> 6);
    if (e >= EP_EDGES) return;
    int s, d; edge_sd(ei, e, s, d);
    const float4 p   = ((const float4*)pbuf)[e];
    const float4 den = ((const float4*)segsum)[d];
    const float a0 = p.x / (den.x + 1e-16f);
    const float a1 = p.y / (den.y + 1e-16f);
    const float a2 = p.z / (den.z + 1e-16f);
    const float a3 = p.w / (den.w + 1e-16f);
    const float* xr = xh + (size_t)s * HC + c;
    const float acc = a0 * xr[0] + a1 * xr[64] + a2 * xr[128] + a3 * xr[192];
    atomicAdd(&y[(size_t)d * C_OUT + c], 0.25f * acc);
}

// ---------------------------------------------------------------- bias + BN stats
// in-place y += b[c]; accumulate per-channel sum / sumsq. 64 nodes per block.
__global__ void stats_kernel(float* __restrict__ y,
                             const float* __restrict__ bias,
                             float* __restrict__ chanSum,
                             float* __restrict__ chanSq) {
    __shared__ float sS[256];
    __shared__ float sQ[256];
    const int t = threadIdx.x;
    const int c = t & 63;
    const int r = t >> 6;                    // 0..3
    const float bc = bias[c];
    float ls = 0.0f, lq = 0.0f;
    #pragma unroll
    for (int i = 0; i < 16; ++i) {
        const int n = blockIdx.x * 64 + i * 4 + r;
        if (n < N_NODES) {
            float v = y[(size_t)n * C_OUT + c] + bc;
            y[(size_t)n * C_OUT + c] = v;
            ls += v; lq += v * v;
        }
    }
    sS[t] = ls; sQ[t] = lq;
    __syncthreads();
    if (t < 64) {
        atomicAdd(&chanSum[t], sS[t] + sS[t + 64] + sS[t + 128] + sS[t + 192]);
        atomicAdd(&chanSq[t],  sQ[t] + sQ[t + 64] + sQ[t + 128] + sQ[t + 192]);
    }
}

__global__ void bn_relu_kernel(const float* __restrict__ y,
                               const float* __restrict__ gamma,
                               const float* __restrict__ beta,
                               const float* __restrict__ chanSum,
                               const float* __restrict__ chanSq,
                               float* __restrict__ out) {
    const int i = blockIdx.x * blockDim.x + threadIdx.x;
    if (i >= N_NODES * C_OUT) return;
    const int c = i & 63;
    const float invN = 1.0f / (float)N_NODES;
    const float mu  = chanSum[c] * invN;
    const float var = chanSq[c] * invN - mu * mu;
    const float v = gamma[c] * (y[i] - mu) * rsqrtf(var + BN_EPS) + beta[c];
    out[i] = v > 0.0f ? v : 0.0f;
}

// ---------------------------------------------------------------- driver
extern "C" void kernel_launch(void* const* d_in, const int* in_sizes, int n_in,
                              void* d_out, int out_size, void* d_ws, size_t ws_size,
                              hipStream_t stream) {
    (void)in_sizes; (void)n_in; (void)out_size; (void)ws_size;
    const float* x   = (const float*)d_in[0];
    const int*   ei  = (const int*)d_in[1];
    // d_in[2] edge_weight is unused (edge_dim=None in reference)
    const float* prm[2][6];
    for (int l = 0; l < 2; ++l)
        for (int j = 0; j < 6; ++j)
            prm[l][j] = (const float*)d_in[3 + l * 6 + j];  // W, asrc, adst, b, gamma, beta

    float* ws = (float*)d_ws;
    float* xh      = ws;                          // N*256
    float* alsrc   = xh + (size_t)N_NODES * HC;   // N*4
    float* aldst   = alsrc + N_NODES * HEADS;     // N*4
    float* segmax  = aldst + N_NODES * HEADS;     // N*4
    float* segsum  = segmax + N_NODES * HEADS;    // N*4
    float* pbuf    = segsum + N_NODES * HEADS;    // EP*4
    float* y0      = pbuf + (size_t)EP_EDGES * HEADS; // N*64 (layer-1 activation, BN in place)
    float* chanSum = y0 + (size_t)N_NODES * C_OUT;    // 64
    float* chanSq  = chanSum + 64;                    // 64

    float* out_final = (float*)d_out;                 // N*64
    float* out_saved = out_final + (size_t)N_NODES * C_OUT; // N*64 (layer-2 pre-BN)

    const int NV   = N_NODES * HEADS;
    const int NY   = N_NODES * C_OUT;
    const int TB   = 256;

    for (int l = 0; l < 2; ++l) {
        const float* xin  = (l == 0) ? x  : y0;
        float*       ybuf = (l == 0) ? y0 : out_saved;
        float*       dest = (l == 0) ? y0 : out_final;

        fill_kernel<<<(NV + TB - 1) / TB, TB, 0, stream>>>(segmax, -INFINITY, NV);
        fill_kernel<<<(NV + TB - 1) / TB, TB, 0, stream>>>(segsum, 0.0f, NV);
        fill_kernel<<<(NY + TB - 1) / TB, TB, 0, stream>>>(ybuf, 0.0f, NY);
        fill_kernel<<<1, 128, 0, stream>>>(chanSum, 0.0f, 128);  // chanSum+chanSq contiguous

        gemm_wmma_kernel<<<dim3(N_NODES / 16, 2), TB, 0, stream>>>(xin, prm[l][0], xh);
        al_kernel<<<N_NODES, TB, 0, stream>>>(xh, prm[l][1], prm[l][2], alsrc, aldst);
        edge_max_kernel<<<(EP_EDGES + TB - 1) / TB, TB, 0, stream>>>(ei, alsrc, aldst, segmax);
        edge_exp_kernel<<<(EP_EDGES + TB - 1) / TB, TB, 0, stream>>>(ei, alsrc, aldst, segmax,
                                                                     pbuf, segsum);
        edge_agg_kernel<<<(EP_EDGES + 3) / 4, TB, 0, stream>>>(ei, pbuf, segsum, xh, ybuf);
        stats_kernel<<<(N_NODES + 63) / 64, TB, 0, stream>>>(ybuf, prm[l][3], chanSum, chanSq);
        bn_relu_kernel<<<(NY + TB - 1) / TB, TB, 0, stream>>>(ybuf, prm[l][4], prm[l][5],
                                                              chanSum, chanSq, dest);
    }
}